// FilterBankKernel_15212774162740
// MI455X (gfx1250) — compile-verified
//
#include <hip/hip_runtime.h>
#include <hip/hip_bf16.h>

typedef __attribute__((ext_vector_type(2))) float v2f;
typedef __attribute__((ext_vector_type(8))) float v8f;

#define BATCH 256
#define NFILT 64
#define NTOT  147456   // 128*128*3*3

// ---------------- Phase 1: softmax over the 64-wide filter axis --------------
// One wave32 per row; each lane owns elements (lane, lane+32).
__global__ void softmax_rows(const float* __restrict__ x, float* __restrict__ p) {
    const int row  = blockIdx.x;
    const int lane = threadIdx.x;  // 0..31

    float a = x[row * NFILT + lane];
    float b = x[row * NFILT + lane + 32];

    float m = fmaxf(a, b);
    #pragma unroll
    for (int off = 16; off > 0; off >>= 1)
        m = fmaxf(m, __shfl_xor(m, off, 32));

    float ea = __expf(a - m);
    float eb = __expf(b - m);
    float s  = ea + eb;
    #pragma unroll
    for (int off = 16; off > 0; off >>= 1)
        s += __shfl_xor(s, off, 32);

    float inv = __frcp_rn(s);
    p[row * NFILT + lane]      = ea * inv;
    p[row * NFILT + lane + 32] = eb * inv;
}

// ---------------- Phase 2: 256x147456 = probs(256x64) @ W(64x147456) ---------
// One wave per 16-wide N strip. B block (64x16) lives in 32 VGPRs for the
// whole strip; loop over all 16 M tiles reusing it, so W is read exactly once.
__global__ void __launch_bounds__(256)
bank_gemm(const float* __restrict__ probs,   // (256, 64) row-major
          const float* __restrict__ W,       // (64, 147456) row-major
          float* __restrict__ out)           // (256, 147456) row-major
{
    const int wave  = (blockIdx.x * blockDim.x + threadIdx.x) >> 5;  // N-strip id
    const int lane  = threadIdx.x & 31;
    const int half  = lane >> 4;    // 0: lanes 0-15, 1: lanes 16-31
    const int nl    = lane & 15;    // column within the 16-wide strip

    const int n0 = wave * 16;       // max 9215*16 < 2^31, fine

    // ---- Load B fragments: 16 k-slices of the 64x16 weight block -----------
    // Layout (assumed, mirrors A transposed): VGPR0 lanes0-15 = row K=k0,
    // lanes16-31 = row K=k0+1; VGPR1 = rows k0+2 / k0+3.
    v2f bfrag[16];
    #pragma unroll
    for (int ks = 0; ks < 16; ++ks) {
        const int k0 = ks * 4;
        const float* wp = W + (size_t)(k0 + half) * NTOT + n0 + nl;
        bfrag[ks].x = wp[0];
        bfrag[ks].y = wp[2 * NTOT];
    }

    // ---- Sweep all 16 M tiles (batch rows 0..255) reusing B ----------------
    #pragma unroll 1
    for (int mt = 0; mt < 16; ++mt) {
        v8f acc = {};
        #pragma unroll
        for (int ks = 0; ks < 16; ++ks) {
            const int k0 = ks * 4;
            // A 16x4 fragment: lanes0-15 hold (K=k0,k0+1), lanes16-31 (k0+2,k0+3)
            const float* ap = probs + (mt * 16 + nl) * NFILT + k0 + half * 2;
            v2f afrag;
            afrag.x = ap[0];
            afrag.y = ap[1];
            acc = __builtin_amdgcn_wmma_f32_16x16x4_f32(
                /*neg_a=*/false, afrag,
                /*neg_b=*/false, bfrag[ks],
                /*c_mod=*/(short)0, acc,
                /*reuse_a=*/false, /*reuse_b=*/false);
        }
        // ---- Store 16x16 D tile: VGPR r -> row (mt*16 + r + half*8) --------
        #pragma unroll
        for (int r = 0; r < 8; ++r) {
            const size_t row = (size_t)(mt * 16 + r + half * 8);
            __builtin_nontemporal_store(acc[r], out + row * NTOT + n0 + nl);
        }
    }
}

extern "C" void kernel_launch(void* const* d_in, const int* in_sizes, int n_in,
                              void* d_out, int out_size, void* d_ws, size_t ws_size,
                              hipStream_t stream) {
    const float* bank_request = (const float*)d_in[0];   // (256, 64)
    const float* weight       = (const float*)d_in[1];   // (64, 128,128,3,3) = (64, 147456)
    float*       out          = (float*)d_out;           // (256, 147456)
    float*       probs        = (float*)d_ws;            // 256*64 floats = 64 KB scratch

    // Phase 1: softmax -> probs
    softmax_rows<<<BATCH, 32, 0, stream>>>(bank_request, probs);

    // Phase 2: GEMM. 147456/16 = 9216 waves; 8 waves (256 thr) per block.
    const int nwaves = NTOT / 16;          // 9216
    const int block  = 256;
    const int grid   = nwaves / (block / 32);  // 1152
    bank_gemm<<<grid, block, 0, stream>>>(probs, weight, out);
}